// LogSig_var_5111011082838
// MI455X (gfx1250) — compile-verified
//
#include <hip/hip_runtime.h>

// LogSig depth-2 on gfx1250: one wave32 per (batch,segment), segment GEMM via
// V_WMMA_F32_16X16X4_F32 with a scalar (wave-uniform) K-loop, tail handled by
// value-level predication (EXEC stays all-ones for WMMA), antisymmetrization
// through a wave-private LDS tile synchronized with s_wait_dscnt only.

typedef __attribute__((ext_vector_type(2))) float v2f;
typedef __attribute__((ext_vector_type(8))) float v8f;

#define IN_CH 16
#define NSEG  32
#define NT    2048
#define NOUT  136   // 16 (lvl1) + 120 (triu lvl2)

__global__ __launch_bounds__(256)
void logsig_wmma_kernel(const float* __restrict__ inp,
                        const int*   __restrict__ length,
                        float*       __restrict__ out)
{
    __shared__ float lds[8 * 256];          // 8 waves * 16x16 G tile (wave-private)

    const int lane = threadIdx.x & 31;
    const int wv   = threadIdx.x >> 5;
    const int task = blockIdx.x * 8 + wv;   // 0..4095
    const int b    = task >> 5;             // batch row 0..127
    const int s    = task & 31;             // segment 0..31
    const int bb   = b >> 4;                // length index (batch_size=16)

    // Segment boundaries: match jnp.round (round-half-even) via rintf.
    // Force wave-uniform (SGPR) so the K-loop is a scalar branch loop.
    const float Lf = (float)length[bb];
    const int ta = __builtin_amdgcn_readfirstlane(
        (int)rintf(1.0f + (Lf - 1.0f) * ((float)s       * (1.0f/32.0f))) - 1);
    const int te = __builtin_amdgcn_readfirstlane(
        (int)rintf(1.0f + (Lf - 1.0f) * ((float)(s + 1) * (1.0f/32.0f))) - 1);

    const int c     = lane & 15;            // channel / column handled by this lane
    const int h     = (lane >= 16) ? 2 : 0; // K-half offset for A/B fragments
    const int rbase = (lane >= 16) ? 8 : 0; // C/D row base for this lane

    const float* xcol = inp + ((size_t)b * NT) * IN_CH + c;   // x[t][c] at xcol[t*16]

    // M[i][j] = sum_t x_t[i] * dx_t[j]  == (16 x K) * (K x 16), K = te-ta
    v8f acc = {0.f, 0.f, 0.f, 0.f, 0.f, 0.f, 0.f, 0.f};

    // ---- main loop: full groups of 4 timesteps, no clamping, no predication ----
    const float* p = xcol + (size_t)(ta + h) * IN_CH;
    int t0 = ta;
    for (; t0 + 4 <= te; t0 += 4) {
        const float x0 = p[0 * IN_CH];      // x[t0+h]  [c]
        const float x1 = p[1 * IN_CH];      // x[t0+h+1][c]
        const float x2 = p[2 * IN_CH];      // x[t0+h+2][c]  (index <= te <= 2047)
        v2f a, bf;
        a.x  = x0;          a.y  = x1;
        bf.x = x1 - x0;     bf.y = x2 - x1;
        acc = __builtin_amdgcn_wmma_f32_16x16x4_f32(
                  false, a, false, bf, (short)0, acc, false, false);
        p += 4 * IN_CH;
    }

    // ---- tail: 1..3 remaining timesteps, value-level predication only ----
    if (t0 < te) {                          // scalar condition -> EXEC untouched
        const int tA0 = t0 + h;
        const int tA1 = tA0 + 1;
        const int m0 = (tA0     < NT - 1) ? tA0     : NT - 1;
        const int m1 = (tA1     < NT - 1) ? tA1     : NT - 1;
        const int m2 = (tA1 + 1 < NT - 1) ? tA1 + 1 : NT - 1;
        const float x0 = xcol[(size_t)m0 * IN_CH];
        const float x1 = xcol[(size_t)m1 * IN_CH];
        const float x2 = xcol[(size_t)m2 * IN_CH];
        const bool ok0 = tA0 < te;
        const bool ok1 = tA1 < te;
        v2f a, bf;
        a.x  = ok0 ? x0 : 0.0f;
        a.y  = ok1 ? x1 : 0.0f;
        bf.x = ok0 ? (x1 - x0) : 0.0f;
        bf.y = ok1 ? (x2 - x1) : 0.0f;
        acc = __builtin_amdgcn_wmma_f32_16x16x4_f32(
                  false, a, false, bf, (short)0, acc, false, false);
    }

    // lvl1 for this lane's channel
    const float xa_c   = xcol[(size_t)ta * IN_CH];
    const float xe_c   = xcol[(size_t)te * IN_CH];
    const float lvl1_c = xe_c - xa_c;

    // G[i][j] = M[i][j] - xa[i]*lvl1[j]; this lane owns column j=c, rows rbase..rbase+7
    const float* xarow = inp + (((size_t)b * NT) + ta) * IN_CH + rbase;
    float* ldsw = lds + wv * 256;
#pragma unroll
    for (int v = 0; v < 8; ++v) {
        const float g = acc[v] - xarow[v] * lvl1_c;
        ldsw[(rbase + v) * 16 + c] = g;
    }
    // Tile is wave-private; LDS ops are in-order per wave. Fence the compiler and
    // drain DS before the transposed reads -- no block barrier needed.
    asm volatile("s_wait_dscnt 0x0" ::: "memory");

    float* orow = out + (size_t)task * NOUT;
    if (lane < 16) orow[c] = lvl1_c;

    // lvl2: A = 0.5*(G - G^T), row-major upper triangle (triu k=1)
    for (int k = lane; k < 120; k += 32) {
        int kk = k, i = 0;
        while (kk >= 15 - i) { kk -= 15 - i; ++i; }
        const int j = i + 1 + kk;
        orow[16 + k] = 0.5f * (ldsw[i * 16 + j] - ldsw[j * 16 + i]);
    }
}

extern "C" void kernel_launch(void* const* d_in, const int* in_sizes, int n_in,
                              void* d_out, int out_size, void* d_ws, size_t ws_size,
                              hipStream_t stream)
{
    (void)in_sizes; (void)n_in; (void)out_size; (void)d_ws; (void)ws_size;
    const float* inp    = (const float*)d_in[0];
    const int*   length = (const int*)d_in[1];
    float*       out    = (float*)d_out;

    // 128 batch rows * 32 segments = 4096 wave-tasks; 8 waves per 256-thread block
    dim3 grid(512), block(256);
    hipLaunchKernelGGL(logsig_wmma_kernel, grid, block, 0, stream, inp, length, out);
}